// CausalConv1d_59657095741678
// MI455X (gfx1250) — compile-verified
//
#include <hip/hip_runtime.h>

// Causal depthwise conv1d: out[b,t,f] = sum_{k=0..3} w[k,f] * x[b, t-3+k, f] + bias[f]
// Shapes: x (B,T,F) fp32, w (K,1,F) fp32, b (F,) fp32, out (B,T,F) fp32.
// Memory-bound (1 FLOP/byte): pure 128-bit streaming kernel with register
// sliding window, NT cache hints, and gfx1250 global_prefetch_b8 look-ahead.

typedef float vf4 __attribute__((ext_vector_type(4)));

constexpr int Bc = 4;
constexpr int Tc = 8192;
constexpr int Fc = 1024;
constexpr int C4 = Fc / 4;   // 256 float4 feature columns
constexpr int TT = 64;       // time steps per block (halo overhead 3/64)
constexpr int THREADS = C4;  // one block covers all features of one (b, t-chunk)

__global__ __launch_bounds__(THREADS)
void CausalConv1d_59657095741678_kernel(const float* __restrict__ xp,
                                        const float* __restrict__ wp,
                                        const float* __restrict__ bp,
                                        float* __restrict__ op) {
    const int c = threadIdx.x;                 // float4 feature column 0..255
    const int nchunks = Tc / TT;               // 128
    const int b  = blockIdx.x / nchunks;
    const int t0 = (blockIdx.x % nchunks) * TT;

    const vf4* __restrict__ x4 = (const vf4*)xp;
    const vf4* __restrict__ w4 = (const vf4*)wp;
    const vf4* __restrict__ b4 = (const vf4*)bp;
    vf4* __restrict__ o4 = (vf4*)op;

    // Per-feature taps (K=4) and bias, loaded once (hits L2/WGP$ across blocks).
    const vf4 w0 = w4[0 * C4 + c];
    const vf4 w1 = w4[1 * C4 + c];
    const vf4 w2 = w4[2 * C4 + c];
    const vf4 w3 = w4[3 * C4 + c];
    const vf4 bias = b4[c];

    const long row0 = (long)b * Tc;

    vf4 zero = 0.0f;
    // Sliding-window history: h0 = x[t0-3], h1 = x[t0-2], h2 = x[t0-1] (zero-padded).
    vf4 h0 = (t0 >= 3) ? __builtin_nontemporal_load(&x4[(row0 + t0 - 3) * C4 + c]) : zero;
    vf4 h1 = (t0 >= 2) ? __builtin_nontemporal_load(&x4[(row0 + t0 - 2) * C4 + c]) : zero;
    vf4 h2 = (t0 >= 1) ? __builtin_nontemporal_load(&x4[(row0 + t0 - 1) * C4 + c]) : zero;

    long idx = (row0 + t0) * C4 + c;
    for (int ii = 0; ii < TT; ii += 8) {
        // gfx1250 speculative prefetch ~32 rows (128 KB of x) ahead; OOB at the
        // tail is silently dropped per ISA prefetch semantics.
        __builtin_prefetch(&x4[idx + 32 * (long)C4], 0, 0);
        #pragma unroll
        for (int i = 0; i < 8; ++i) {
            vf4 cur = __builtin_nontemporal_load(&x4[idx]);     // read-once stream
            vf4 r = ((w0 * h0 + w1 * h1) + (w2 * h2 + w3 * cur)) + bias;
            __builtin_nontemporal_store(r, &o4[idx]);           // write-once stream
            h0 = h1; h1 = h2; h2 = cur;
            idx += C4;
        }
    }
}

extern "C" void kernel_launch(void* const* d_in, const int* in_sizes, int n_in,
                              void* d_out, int out_size, void* d_ws, size_t ws_size,
                              hipStream_t stream) {
    const float* x = (const float*)d_in[0];   // (B,T,F)
    const float* w = (const float*)d_in[1];   // (K,1,F)
    const float* b = (const float*)d_in[2];   // (F,)
    float* out = (float*)d_out;               // (B,T,F)

    dim3 grid(Bc * (Tc / TT));                // 512 blocks
    dim3 block(THREADS);                      // 256 threads = 8 wave32s
    CausalConv1d_59657095741678_kernel<<<grid, block, 0, stream>>>(x, w, b, out);
}